// OccamNet_38079180046536
// MI455X (gfx1250) — compile-verified
//
#include <hip/hip_runtime.h>
#include <math.h>

// OccamNet path-probability evaluation for MI455X (gfx1250, wave32).
//
//   (1) lse_all_kernel: row-wise logsumexp of w0/w1/w2/w_out, all four
//       matrices in ONE launch (385 row-tiles -> 385 waves in flight).
//       Streaming-bound: w1+w2 = 33.5MB; second pass re-reads from L2
//       (192MB). Sum-of-exp uses V_WMMA_F32_16X16X4_F32 with an all-ones
//       B matrix: one wave per 16 rows, lane l supplies
//       A[l%16, 2*(l>>4)+{0,1}] per the CDNA5 f32 A layout; two D
//       accumulators break the WMMA RAW chain.
//       exp arguments are w - rowmax in [-~8, 0] (no overflow/underflow
//       possible), so native v_exp_f32 (__expf) is exact enough (~1e-7
//       abs error on the resulting lse) and avoids the ~12-VALU libm
//       special-case path per element.
//   (2) path_kernel: one thread per (sample, output) walks the sampled
//       path backwards with 4 dependent gathers (65K threads; accurate
//       sinf/expf kept here since the count is negligible).

typedef float v2f __attribute__((ext_vector_type(2)));
typedef float v8f __attribute__((ext_vector_type(8)));

#define S_ 8192
#define D_ 64
#define N_ 2048
#define O_ 8

#define TILES_W  (N_ / 16)      // 128 row-tiles for w0/w1/w2
#define TILES_ALL (3 * TILES_W + 1)

// One wave32 per 16 rows of one matrix. ncols must be a multiple of 8.
__global__ __launch_bounds__(32) void lse_all_kernel(
    const float* __restrict__ w0p, const float* __restrict__ w1p,
    const float* __restrict__ w2p, const float* __restrict__ woutp,
    float* __restrict__ l0, float* __restrict__ l1,
    float* __restrict__ l2, float* __restrict__ lO)
{
    // ---- select matrix by block (scalar branch; EXEC untouched) ----
    const int b = blockIdx.x;
    const float* W;
    float* lse;
    int nrows, ncols, tile;
    if (b < TILES_W)            { W = w0p;   lse = l0; nrows = N_; ncols = D_; tile = b; }
    else if (b < 2 * TILES_W)   { W = w1p;   lse = l1; nrows = N_; ncols = N_; tile = b - TILES_W; }
    else if (b < 3 * TILES_W)   { W = w2p;   lse = l2; nrows = N_; ncols = N_; tile = b - 2 * TILES_W; }
    else                        { W = woutp; lse = lO; nrows = O_; ncols = N_; tile = 0; }

    const int lane = threadIdx.x;     // 0..31
    const int hf   = lane >> 4;       // half-wave: supplies K={0,1} or K={2,3}
    const int m    = lane & 15;       // row within the 16-row tile
    const int row  = tile * 16 + m;
    const int rowc = (row < nrows) ? row : (nrows - 1);   // clamp (w_out tail), no branch
    const float* rp = W + (size_t)rowc * (size_t)ncols;

    const int iters = ncols >> 3;     // 8 columns per iteration per row

    // ---- pass 1: per-row max (float4 streaming; lanes m and m+16 split cols) ----
    float mx = -3.402823466e38f;
    {
        const float4* p = (const float4*)(rp + 4 * hf);
        for (int it = 0; it < iters; ++it) {
            float4 v = p[2 * it];     // stride 8 floats
            mx = fmaxf(mx, fmaxf(fmaxf(v.x, v.y), fmaxf(v.z, v.w)));
        }
    }
    // lanes m and m+16 covered complementary column sets of the same row
    mx = fmaxf(mx, __shfl_xor(mx, 16));

    // ---- pass 2: sum of exp(w - max) via v_wmma_f32_16x16x4_f32, B = ones ----
    // Per WMMA: D[m,n] += A[m,0]+A[m,1]+A[m,2]+A[m,3] (row sums, replicated over n).
    // Two accumulators break the WMMA->WMMA RAW dependence.
    v8f acc0 = {};
    v8f acc1 = {};
    v2f ones; ones.x = 1.0f; ones.y = 1.0f;
    {
        const float4* p = (const float4*)(rp + 4 * hf);
        for (int it = 0; it < iters; ++it) {
            float4 v = p[2 * it];
            v2f e0, e1;
            e0.x = __expf(v.x - mx); e0.y = __expf(v.y - mx);
            e1.x = __expf(v.z - mx); e1.y = __expf(v.w - mx);
            // 8 args: (neg_a, A, neg_b, B, c_mod, C, reuse_a, reuse_b)
            acc0 = __builtin_amdgcn_wmma_f32_16x16x4_f32(false, e0, false, ones,
                                                         (short)0, acc0, false, false);
            acc1 = __builtin_amdgcn_wmma_f32_16x16x4_f32(false, e1, false, ones,
                                                         (short)0, acc1, false, false);
        }
    }
    v8f acc = acc0 + acc1;   // same replicated-row-sum layout -> elementwise add is exact

    // D layout (16x16 f32): VGPR j holds row j (lanes 0-15) / row j+8 (lanes 16-31).
    __shared__ float ssum[16];
    if (lane == 0) {
        ssum[0] = acc[0]; ssum[1] = acc[1]; ssum[2] = acc[2]; ssum[3] = acc[3];
        ssum[4] = acc[4]; ssum[5] = acc[5]; ssum[6] = acc[6]; ssum[7] = acc[7];
    }
    if (lane == 16) {
        ssum[8]  = acc[0]; ssum[9]  = acc[1]; ssum[10] = acc[2]; ssum[11] = acc[3];
        ssum[12] = acc[4]; ssum[13] = acc[5]; ssum[14] = acc[6]; ssum[15] = acc[7];
    }
    __syncthreads();
    if (lane < 16 && row < nrows) {
        // lane == m == tile row; mx at lane m is the full max of row m
        lse[row] = mx + __logf(ssum[lane]);
    }
}

// One thread per (sample, output): walk the sampled path backwards.
__global__ __launch_bounds__(256) void path_kernel(
    const float* __restrict__ x,    const float* __restrict__ w0,
    const float* __restrict__ w1,   const float* __restrict__ w2,
    const float* __restrict__ wout,
    const int* __restrict__ c0,     const int* __restrict__ c1,
    const int* __restrict__ c2,     const int* __restrict__ keys_out,
    const float* __restrict__ lse0, const float* __restrict__ lse1,
    const float* __restrict__ lse2, const float* __restrict__ lseO,
    float* __restrict__ out)
{
    const int idx = blockIdx.x * blockDim.x + threadIdx.x;   // over S*O
    if (idx >= S_ * O_) return;
    const int s = idx >> 3;          // idx / O_
    const int o = idx & (O_ - 1);    // idx % O_

    const int r2 = keys_out[idx];
    float logp = wout[o * N_ + r2] - lseO[o];

    const int r1 = c2[(size_t)s * N_ + r2];
    logp += w2[(size_t)r2 * N_ + r1] - lse2[r2];

    const int r0 = c1[(size_t)s * N_ + r1];
    logp += w1[(size_t)r1 * N_ + r0] - lse1[r1];

    const int cin = c0[(size_t)s * N_ + r0];
    logp += w0[r0 * D_ + cin] - lse0[r0];

    out[idx]           = sinf(sinf(sinf(x[s * D_ + cin])));   // y
    out[S_ * O_ + idx] = expf(logp);                          // path probability
}

extern "C" void kernel_launch(void* const* d_in, const int* in_sizes, int n_in,
                              void* d_out, int out_size, void* d_ws, size_t ws_size,
                              hipStream_t stream)
{
    const float* x    = (const float*)d_in[0];
    const float* w0   = (const float*)d_in[1];
    const float* w1   = (const float*)d_in[2];
    const float* w2   = (const float*)d_in[3];
    const float* wout = (const float*)d_in[4];
    const int*   c0   = (const int*)d_in[5];
    const int*   c1   = (const int*)d_in[6];
    const int*   c2   = (const int*)d_in[7];
    const int*   keys = (const int*)d_in[8];
    float* out = (float*)d_out;

    // scratch: 3*2048 + 8 floats of logsumexp values (fully rewritten each call)
    float* lse0 = (float*)d_ws;
    float* lse1 = lse0 + N_;
    float* lse2 = lse1 + N_;
    float* lseO = lse2 + N_;

    // all four matrices' row-logsumexp in one launch (385 waves in flight)
    lse_all_kernel<<<TILES_ALL, 32, 0, stream>>>(w0, w1, w2, wout,
                                                 lse0, lse1, lse2, lseO);

    path_kernel<<<(S_ * O_) / 256, 256, 0, stream>>>(x, w0, w1, w2, wout,
                                                     c0, c1, c2, keys,
                                                     lse0, lse1, lse2, lseO, out);
}